// GATv2Conv_2997887172725
// MI455X (gfx1250) — compile-verified
//
#include <hip/hip_runtime.h>
#include <hip/hip_bf16.h>
#include <math.h>

#define N_NODES 100000
#define N_EDGES 1600000
#define IN_CH   128
#define HEADS   4
#define OUT_CH  32
#define HC      (HEADS * OUT_CH)   // 128

typedef float v2f __attribute__((ext_vector_type(2)));
typedef float v8f __attribute__((ext_vector_type(8)));

// ---------------------------------------------------------------------------
// Kernel A: proj = x @ W using fp32 WMMA (V_WMMA_F32_16X16X4_F32).
// grid.x = N_NODES/16 row strips; block = 256 threads = 8 waves.
// Wave w computes the 16x16 tile (rows rowBase..+15, cols 16w..16w+15).
// K staged via LDS in chunks of 32 (pad stride 36 floats -> conflict-free b64).
// ---------------------------------------------------------------------------
#define XS_STRIDE 36
__global__ __launch_bounds__(256)
void gat_proj_wmma(const float* __restrict__ x, const float* __restrict__ W,
                   float* __restrict__ proj) {
  __shared__ float xs[16 * XS_STRIDE];    // x strip  [row][k-chunk]
  __shared__ float wt[HC * XS_STRIDE];    // W^T chunk [col][k-chunk]

  const int t       = threadIdx.x;
  const int lane    = t & 31;
  const int wave    = t >> 5;
  const int rowBase = blockIdx.x * 16;
  const int colBase = wave * 16;
  const int r       = lane & 15;          // row (A) / col (B) within tile
  const int koff    = (lane >> 4) << 1;   // 0 for lanes 0-15, 2 for lanes 16-31

  v8f acc = {};
  for (int kc = 0; kc < IN_CH; kc += 32) {
    __syncthreads();
    // stage x[rowBase..+15][kc..kc+31] (coalesced: consecutive k per row)
    for (int i = t; i < 16 * 32; i += 256) {
      int rr = i >> 5, kk = i & 31;
      xs[rr * XS_STRIDE + kk] = x[(size_t)(rowBase + rr) * IN_CH + kc + kk];
    }
    // stage W[kc..kc+31][0..127] transposed (coalesced global reads over cols)
    for (int i = t; i < 32 * HC; i += 256) {
      int kk = i >> 7, cc = i & 127;
      wt[cc * XS_STRIDE + kk] = W[(size_t)(kc + kk) * HC + cc];
    }
    __syncthreads();

    // 8 WMMA steps of K=4 each over this chunk.
    // A 16x4 f32 layout (ISA 7.12.2): VGPR0 = K0 (lanes 0-15) / K2 (16-31),
    // VGPR1 = K1 / K3.  B 4x16 mirrors it (rows split lane-halves by K/2).
    #pragma unroll
    for (int k0 = 0; k0 < 32; k0 += 4) {
      v2f a = *(const v2f*)&xs[r * XS_STRIDE + k0 + koff];
      v2f b = *(const v2f*)&wt[(colBase + r) * XS_STRIDE + k0 + koff];
      acc = __builtin_amdgcn_wmma_f32_16x16x4_f32(
          /*neg_a=*/false, a, /*neg_b=*/false, b,
          /*c_mod=*/(short)0, acc, /*reuse_a=*/false, /*reuse_b=*/false);
    }
  }

  // C/D layout: VGPR v -> row v (lanes 0-15) / row v+8 (lanes 16-31), col = lane%16
  const int rOff = (lane >> 4) * 8;
  #pragma unroll
  for (int v = 0; v < 8; ++v)
    proj[(size_t)(rowBase + rOff + v) * HC + colBase + r] = acc[v];
}

// ---------------------------------------------------------------------------
// Monotone uint encoding of float so unsigned atomicMax == float max.
// ---------------------------------------------------------------------------
__device__ __forceinline__ unsigned encodeF(float f) {
  unsigned u = __float_as_uint(f);
  return (u & 0x80000000u) ? ~u : (u | 0x80000000u);
}
__device__ __forceinline__ float decodeF(unsigned m) {
  return __uint_as_float((m & 0x80000000u) ? (m ^ 0x80000000u) : ~m);
}

// ---------------------------------------------------------------------------
// Kernel B: per-edge scores + global per-head max.
// One wave per edge: lane handles 4 channels (float4), 8 lanes per head.
// proj rows (51.2 MB) are L2-resident -> gathers hit L2.
// ---------------------------------------------------------------------------
__global__ __launch_bounds__(256)
void gat_edge_scores(const float* __restrict__ proj, const int* __restrict__ ei,
                     const float* __restrict__ att, float* __restrict__ scores,
                     unsigned* __restrict__ gmaxU) {
  __shared__ unsigned smax[HEADS];
  const int t = threadIdx.x, lane = t & 31, wave = t >> 5;
  if (t < HEADS) smax[t] = 0u;   // 0 == encoding below any real float
  __syncthreads();

  const int   c0 = lane * 4;     // 4 contiguous channels per lane
  const int   h  = lane >> 3;    // head owning these channels
  const float4 av = *(const float4*)&att[c0];

  for (int e = blockIdx.x * 8 + wave; e < N_EDGES; e += gridDim.x * 8) {
    const int row = ei[e];
    const int col = ei[N_EDGES + e];
    const float4 s4 = *(const float4*)&proj[(size_t)row * HC + c0];
    const float4 d4 = *(const float4*)&proj[(size_t)col * HC + c0];
    float s = tanhf(s4.x + d4.x) * av.x
            + tanhf(s4.y + d4.y) * av.y
            + tanhf(s4.z + d4.z) * av.z
            + tanhf(s4.w + d4.w) * av.w;
    // reduce over the 8 lanes of this head
    s += __shfl_xor(s, 1, 32);
    s += __shfl_xor(s, 2, 32);
    s += __shfl_xor(s, 4, 32);
    if ((lane & 7) == 0) {
      scores[(size_t)e * HEADS + h] = s;
      atomicMax(&smax[h], encodeF(s));
    }
  }
  __syncthreads();
  if (t < HEADS) atomicMax(&gmaxU[t], smax[t]);
}

// ---------------------------------------------------------------------------
// Kernel C: w = exp(score - max); scatter-add w*src into out[col], w into norm.
// fp32 global atomics resolve in L2 (out is 51.2 MB, L2-resident).
// ---------------------------------------------------------------------------
__global__ __launch_bounds__(256)
void gat_edge_scatter(const float* __restrict__ proj, const int* __restrict__ ei,
                      const float* __restrict__ scores,
                      const unsigned* __restrict__ gmaxU,
                      float* __restrict__ out, float* __restrict__ norm) {
  __shared__ float mx[HEADS];
  const int t = threadIdx.x, lane = t & 31, wave = t >> 5;
  if (t < HEADS) mx[t] = decodeF(gmaxU[t]);
  __syncthreads();

  const int c0 = lane * 4;
  const int h  = lane >> 3;
  const float m = mx[h];

  for (int e = blockIdx.x * 8 + wave; e < N_EDGES; e += gridDim.x * 8) {
    const int row = ei[e];
    const int col = ei[N_EDGES + e];
    const float w = __expf(scores[(size_t)e * HEADS + h] - m);
    const float4 s4 = *(const float4*)&proj[(size_t)row * HC + c0];
    float* o = &out[(size_t)col * HC + c0];
    atomicAdd(o + 0, s4.x * w);
    atomicAdd(o + 1, s4.y * w);
    atomicAdd(o + 2, s4.z * w);
    atomicAdd(o + 3, s4.w * w);
    if ((lane & 7) == 0) atomicAdd(&norm[(size_t)col * HEADS + h], w);
  }
}

// ---------------------------------------------------------------------------
// Kernel D: out /= max(norm, 1e-12)
// ---------------------------------------------------------------------------
__global__ __launch_bounds__(256)
void gat_normalize(float* __restrict__ out, const float* __restrict__ norm) {
  int i = blockIdx.x * 256 + threadIdx.x;
  if (i >= N_NODES * HC) return;
  int n = i >> 7;
  int h = (i & 127) >> 5;
  out[i] = out[i] / fmaxf(norm[n * HEADS + h], 1e-12f);
}

// ---------------------------------------------------------------------------
extern "C" void kernel_launch(void* const* d_in, const int* in_sizes, int n_in,
                              void* d_out, int out_size, void* d_ws, size_t ws_size,
                              hipStream_t stream) {
  const float* x   = (const float*)d_in[0];
  const int*   ei  = (const int*)d_in[1];   // edge_index [2, E] (harness int32)
  const float* W   = (const float*)d_in[2];
  const float* att = (const float*)d_in[3];
  float* out = (float*)d_out;

  char* ws = (char*)d_ws;
  float* proj = (float*)ws;
  size_t off = (size_t)N_NODES * HC * sizeof(float);
  float* scores = (float*)(ws + off); off += (size_t)N_EDGES * HEADS * sizeof(float);
  float* norm   = (float*)(ws + off); off += (size_t)N_NODES * HEADS * sizeof(float);
  unsigned* gmaxU = (unsigned*)(ws + off);

  // deterministic per-call init (graph-capture-safe)
  hipMemsetAsync(out,  0, (size_t)N_NODES * HC * sizeof(float), stream);
  hipMemsetAsync(norm, 0, (size_t)N_NODES * HEADS * sizeof(float), stream);
  hipMemsetAsync(gmaxU, 0, HEADS * sizeof(unsigned), stream);

  gat_proj_wmma<<<N_NODES / 16, 256, 0, stream>>>(x, W, proj);
  gat_edge_scores<<<4096, 256, 0, stream>>>(proj, ei, att, scores, gmaxU);
  gat_edge_scatter<<<8192, 256, 0, stream>>>(proj, ei, scores, gmaxU, out, norm);
  gat_normalize<<<(N_NODES * HC + 255) / 256, 256, 0, stream>>>(out, norm);
}